// Net_38646115729828
// MI455X (gfx1250) — compile-verified
//
#include <hip/hip_runtime.h>
#include <hip/hip_bf16.h>

typedef float v2f __attribute__((ext_vector_type(2)));
typedef float v8f __attribute__((ext_vector_type(8)));

#define BLOCK 256
#define WPB 8  // waves per block (wave32)

// ---------------- degree / normalization ----------------
__global__ __launch_bounds__(BLOCK) void k_deg_init(float* __restrict__ deg, int N) {
  int i = blockIdx.x * BLOCK + threadIdx.x;
  if (i < N) deg[i] = 1.0f;  // self-loop
}

__global__ __launch_bounds__(BLOCK) void k_deg_accum(const int* __restrict__ dst,
                                                     float* __restrict__ deg, int E) {
  int e = blockIdx.x * BLOCK + threadIdx.x;
  if (e < E) atomicAdd(&deg[dst[e]], 1.0f);
}

__global__ __launch_bounds__(BLOCK) void k_deg_rsqrt(float* __restrict__ deg, int N) {
  int i = blockIdx.x * BLOCK + threadIdx.x;
  if (i < N) deg[i] = rsqrtf(fmaxf(deg[i], 1.0f));  // deg -> dis in place
}

// ---------------- fp32 WMMA GEMM: [N,64] @ [64,64] ----------------
// One wave computes a 16-row x 64-col strip: 16 K-steps x 4 col tiles
// = 64 v_wmma_f32_16x16x4_f32 per wave.
// W is staged in LDS as K-pair-interleaved v2f: ldsW[p*64+c] = {W[2p][c], W[2p+1][c]}
// so each B fragment is one aligned ds_load_b64 straight into a WMMA operand
// pair (no v_dual_mov shuffles, half the DS ops).
__global__ __launch_bounds__(BLOCK) void k_gemm64_wmma(const float* __restrict__ X,
                                                       const float* __restrict__ W,
                                                       const float* __restrict__ bias,
                                                       float* __restrict__ out,
                                                       int N, int fuse_bias_relu) {
  __shared__ v2f ldsW[32 * 64];
  for (int i = threadIdx.x; i < 32 * 64; i += BLOCK) {
    const int p = i >> 6, c = i & 63;
    v2f w;
    w.x = W[(2 * p) * 64 + c];
    w.y = W[(2 * p + 1) * 64 + c];
    ldsW[i] = w;
  }
  __syncthreads();

  const int wave = threadIdx.x >> 5;
  const int lane = threadIdx.x & 31;
  const int m    = lane & 15;   // M index (A) / N index (B,D)
  const int half = lane >> 4;   // lane-half selects K pair / row group
  const int row0 = (blockIdx.x * WPB + wave) << 4;
  if (row0 >= N) return;        // after barrier: safe

  int arow = row0 + m;
  if (arow >= N) arow = N - 1;  // clamp (stores are guarded)
  const float* __restrict__ xrow = X + (size_t)arow * 64;

  v8f acc0 = {}, acc1 = {}, acc2 = {}, acc3 = {};
#pragma unroll
  for (int kk = 0; kk < 16; ++kk) {
    const int k = kk * 4 + half * 2;  // A VGPR0: K = 0/2 by lane half; VGPR1: K+1
    const int p = k >> 1;             // LDS pair row
    v2f a;
    a.x = xrow[k];
    a.y = xrow[k + 1];
    const v2f b0 = ldsW[p * 64 +  0 + m];
    const v2f b1 = ldsW[p * 64 + 16 + m];
    const v2f b2 = ldsW[p * 64 + 32 + m];
    const v2f b3 = ldsW[p * 64 + 48 + m];
    acc0 = __builtin_amdgcn_wmma_f32_16x16x4_f32(false, a, false, b0, (short)0, acc0, false, false);
    acc1 = __builtin_amdgcn_wmma_f32_16x16x4_f32(false, a, false, b1, (short)0, acc1, false, false);
    acc2 = __builtin_amdgcn_wmma_f32_16x16x4_f32(false, a, false, b2, (short)0, acc2, false, false);
    acc3 = __builtin_amdgcn_wmma_f32_16x16x4_f32(false, a, false, b3, (short)0, acc3, false, false);
  }

  float bv0 = 0.f, bv1 = 0.f, bv2 = 0.f, bv3 = 0.f;
  if (fuse_bias_relu) {
    bv0 = bias[ 0 + m]; bv1 = bias[16 + m]; bv2 = bias[32 + m]; bv3 = bias[48 + m];
  }
#pragma unroll
  for (int r = 0; r < 8; ++r) {
    const int orow = row0 + r + 8 * half;  // D: VGPR r holds M=r (lanes 0-15), M=r+8 (16-31)
    if (orow < N) {
      float* __restrict__ op = out + (size_t)orow * 64;
      float v0 = acc0[r] + bv0, v1 = acc1[r] + bv1, v2 = acc2[r] + bv2, v3 = acc3[r] + bv3;
      if (fuse_bias_relu) {
        v0 = fmaxf(v0, 0.f); v1 = fmaxf(v1, 0.f); v2 = fmaxf(v2, 0.f); v3 = fmaxf(v3, 0.f);
      }
      op[ 0 + m] = v0; op[16 + m] = v1; op[32 + m] = v2; op[48 + m] = v3;
    }
  }
}

// ---------------- graph aggregation ----------------
__global__ __launch_bounds__(BLOCK) void k_selfloop(const float* __restrict__ h,
                                                    const float* __restrict__ dis,
                                                    float* __restrict__ agg, int N) {
  int gid = blockIdx.x * BLOCK + threadIdx.x;  // N*64 = 6.4M fits int
  int i = gid >> 6;
  if (i < N) {
    float d = dis[i];
    agg[gid] = h[gid] * d * d;  // also initializes agg (poisoned workspace)
  }
}

// 16 threads per edge: coalesced float4 load of h[src], 4 f32 atomics into agg[dst].
// agg (25.6 MB) fits in the 192 MB L2, so atomics resolve at L2 atomic units.
__global__ __launch_bounds__(BLOCK) void k_scatter(const float* __restrict__ h,
                                                   const int* __restrict__ src,
                                                   const int* __restrict__ dst,
                                                   const float* __restrict__ dis,
                                                   float* __restrict__ agg, int E) {
  unsigned gid = blockIdx.x * BLOCK + threadIdx.x;
  unsigned e = gid >> 4;
  if (e >= (unsigned)E) return;
  const int c = (gid & 15) << 2;
  const int s = src[e];
  const int d = dst[e];
  const float w = dis[s] * dis[d];
  const float4 hv = *reinterpret_cast<const float4*>(h + (size_t)s * 64 + c);
  float* __restrict__ ap = agg + (size_t)d * 64 + c;
  atomicAdd(ap + 0, hv.x * w);
  atomicAdd(ap + 1, hv.y * w);
  atomicAdd(ap + 2, hv.z * w);
  atomicAdd(ap + 3, hv.w * w);
}

__global__ __launch_bounds__(BLOCK) void k_bias_relu(float* __restrict__ y,
                                                     const float* __restrict__ b, int N) {
  int gid = blockIdx.x * BLOCK + threadIdx.x;
  if (gid < N * 64) y[gid] = fmaxf(y[gid] + b[gid & 63], 0.0f);
}

// ---------------- final GEMM [N,64]@[64,16] + bias + log_softmax ----------------
// Each 16-lane half of the D fragment holds one row's 16 class logits ->
// log-softmax is a width-16 shuffle reduction, fully in registers.
__global__ __launch_bounds__(BLOCK) void k_final_lsm(const float* __restrict__ X,
                                                     const float* __restrict__ W,
                                                     const float* __restrict__ bias,
                                                     float* __restrict__ out, int N) {
  __shared__ v2f ldsW[32 * 16];  // [pair p][class c] = {W[2p][c], W[2p+1][c]}
  for (int i = threadIdx.x; i < 32 * 16; i += BLOCK) {
    const int p = i >> 4, c = i & 15;
    v2f w;
    w.x = W[(2 * p) * 16 + c];
    w.y = W[(2 * p + 1) * 16 + c];
    ldsW[i] = w;
  }
  __syncthreads();

  const int wave = threadIdx.x >> 5;
  const int lane = threadIdx.x & 31;
  const int m    = lane & 15;
  const int half = lane >> 4;
  const int row0 = (blockIdx.x * WPB + wave) << 4;
  if (row0 >= N) return;

  int arow = row0 + m;
  if (arow >= N) arow = N - 1;
  const float* __restrict__ xrow = X + (size_t)arow * 64;

  v8f acc = {};
#pragma unroll
  for (int kk = 0; kk < 16; ++kk) {
    const int k = kk * 4 + half * 2;
    const int p = k >> 1;
    v2f a;
    a.x = xrow[k];
    a.y = xrow[k + 1];
    const v2f b = ldsW[p * 16 + m];
    acc = __builtin_amdgcn_wmma_f32_16x16x4_f32(false, a, false, b, (short)0, acc, false, false);
  }

  const float bm = bias[m];
#pragma unroll
  for (int r = 0; r < 8; ++r) {
    float v = acc[r] + bm;  // logit for row (row0+r+8*half), class m
    float mx = v;
#pragma unroll
    for (int off = 8; off >= 1; off >>= 1) mx = fmaxf(mx, __shfl_xor(mx, off, 16));
    float ex = __expf(v - mx);
    float s = ex;
#pragma unroll
    for (int off = 8; off >= 1; off >>= 1) s += __shfl_xor(s, off, 16);
    const float lsm = v - mx - __logf(s);
    const int orow = row0 + r + 8 * half;
    if (orow < N) out[(size_t)orow * 16 + m] = lsm;
  }
}

// ---------------- launcher ----------------
extern "C" void kernel_launch(void* const* d_in, const int* in_sizes, int n_in,
                              void* d_out, int out_size, void* d_ws, size_t ws_size,
                              hipStream_t stream) {
  const float* x   = (const float*)d_in[0];
  const int*   ei  = (const int*)d_in[1];
  const float* W1  = (const float*)d_in[2];
  const float* b1  = (const float*)d_in[3];
  const float* W2  = (const float*)d_in[4];
  const float* b2  = (const float*)d_in[5];
  const float* W3  = (const float*)d_in[6];
  const float* b3  = (const float*)d_in[7];
  const float* Wf1 = (const float*)d_in[8];
  const float* bf1 = (const float*)d_in[9];
  const float* Wf2 = (const float*)d_in[10];
  const float* bf2 = (const float*)d_in[11];
  const float* Wf3 = (const float*)d_in[12];
  const float* bf3 = (const float*)d_in[13];

  const int N = in_sizes[0] / 64;
  const int E = in_sizes[1] / 2;
  const int* src = ei;       // edge_index[0]
  const int* dst = ei + E;   // edge_index[1]

  float* dis  = (float*)d_ws;               // N
  float* bufA = dis + N;                    // N*64
  float* bufB = bufA + (size_t)N * 64;      // N*64

  const dim3 blk(BLOCK);
  const int gN       = (N + BLOCK - 1) / BLOCK;
  const int gE       = (E + BLOCK - 1) / BLOCK;
  const int gNH      = (int)(((size_t)N * 64 + BLOCK - 1) / BLOCK);
  const int gScatter = (int)(((size_t)E * 16 + BLOCK - 1) / BLOCK);
  const int rowTiles = (N + 15) / 16;
  const int gGemm    = (rowTiles + WPB - 1) / WPB;

  // symmetric GCN normalization (with self-loops)
  k_deg_init <<<gN, blk, 0, stream>>>(dis, N);
  k_deg_accum<<<gE, blk, 0, stream>>>(dst, dis, E);
  k_deg_rsqrt<<<gN, blk, 0, stream>>>(dis, N);

  const float* Ws[3] = {W1, W2, W3};
  const float* bs[3] = {b1, b2, b3};
  const float* y = x;
  for (int l = 0; l < 3; ++l) {
    k_gemm64_wmma<<<gGemm, blk, 0, stream>>>(y, Ws[l], nullptr, bufA, N, 0);
    k_selfloop   <<<gNH, blk, 0, stream>>>(bufA, dis, bufB, N);
    k_scatter    <<<gScatter, blk, 0, stream>>>(bufA, src, dst, dis, bufB, E);
    k_bias_relu  <<<gNH, blk, 0, stream>>>(bufB, bs[l], N);
    y = bufB;
  }

  k_gemm64_wmma<<<gGemm, blk, 0, stream>>>(bufB, Wf1, bf1, bufA, N, 1);
  k_gemm64_wmma<<<gGemm, blk, 0, stream>>>(bufA, Wf2, bf2, bufB, N, 1);
  k_final_lsm  <<<gGemm, blk, 0, stream>>>(bufB, Wf3, bf3, (float*)d_out, N);
}